// RecyclingEmbedder_44770739093645
// MI455X (gfx1250) — compile-verified
//
#include <hip/hip_runtime.h>
#include <hip/hip_bf16.h>
#include <stdint.h>

#define N_RES 768
#define CM    256
#define CZ    128
#define NBINS 15
#define LN_EPS 1e-5f
#define INF_F  1e8f

// ---- CDNA5 async global->LDS path (proven present on this toolchain) -------
#if defined(__HIP_DEVICE_COMPILE__) && defined(__gfx1250__) && \
    __has_builtin(__builtin_amdgcn_global_load_async_to_lds_b128) && \
    __has_builtin(__builtin_amdgcn_s_wait_asynccnt)
#define USE_ASYNC 1
#else
#define USE_ASYNC 0
#endif

#if USE_ASYNC
typedef int v4i __attribute__((vector_size(16)));
typedef v4i __attribute__((address_space(1)))* g_v4i_ptr;  // "int4 __device__*"
typedef v4i __attribute__((address_space(3)))* l_v4i_ptr;  // "int4 __shared__*"

__device__ __forceinline__ void async_copy16(const float* g, float* l) {
  // Per-lane 16B copy global -> LDS, tracked by ASYNCcnt.
  __builtin_amdgcn_global_load_async_to_lds_b128(
      (g_v4i_ptr)(uintptr_t)g,
      (l_v4i_ptr)(uint32_t)(uintptr_t)l,
      0, 0);
}
#endif

__device__ __forceinline__ float wred(float v) {
#pragma unroll
  for (int off = 16; off > 0; off >>= 1) v += __shfl_xor(v, off, 32);
  return v;
}

// ---------------- m_update: LayerNorm over [768, 256], one wave per row ----
__global__ __launch_bounds__(256) void k_m_layernorm(
    const float* __restrict__ m, const float* __restrict__ g,
    const float* __restrict__ be, float* __restrict__ out) {
  const int gtid = blockIdx.x * 256 + threadIdx.x;
  const int row  = gtid >> 5;
  const int lane = gtid & 31;
  if (row >= N_RES) return;
  const float4* r4 = reinterpret_cast<const float4*>(m + (size_t)row * CM);
  float4 a = r4[lane];
  float4 b = r4[lane + 32];
  float s  = a.x + a.y + a.z + a.w + b.x + b.y + b.z + b.w;
  float ss = a.x*a.x + a.y*a.y + a.z*a.z + a.w*a.w
           + b.x*b.x + b.y*b.y + b.z*b.z + b.w*b.w;
  s = wred(s); ss = wred(ss);
  const float mu  = s * (1.0f / CM);
  const float var = ss * (1.0f / CM) - mu * mu;
  const float inv = rsqrtf(var + LN_EPS);
  const float4* g4 = reinterpret_cast<const float4*>(g);
  const float4* b4 = reinterpret_cast<const float4*>(be);
  float4 ga = g4[lane], gb = g4[lane + 32];
  float4 ba = b4[lane], bb = b4[lane + 32];
  float4 o0, o1;
  o0.x = (a.x - mu) * inv * ga.x + ba.x;
  o0.y = (a.y - mu) * inv * ga.y + ba.y;
  o0.z = (a.z - mu) * inv * ga.z + ba.z;
  o0.w = (a.w - mu) * inv * ga.w + ba.w;
  o1.x = (b.x - mu) * inv * gb.x + bb.x;
  o1.y = (b.y - mu) * inv * gb.y + bb.y;
  o1.z = (b.z - mu) * inv * gb.z + bb.z;
  o1.w = (b.w - mu) * inv * gb.w + bb.w;
  float4* o4 = reinterpret_cast<float4*>(out + (size_t)row * CM);
  o4[lane]      = o0;
  o4[lane + 32] = o1;
}

// ---------------- z_update: streaming LN + distogram LUT add ---------------
// Block = row i (768 blocks x 256 threads = 8 waves). Wave w handles
// j = w, w+8, ... (96 rows). Lane holds channels [4*lane, 4*lane+3].
// LUT folds w_lin + b_lin + be_z; epilogue is 2 packable FMAs/channel.
__global__ __launch_bounds__(256) void k_z_update(
    const float* __restrict__ z, const float* __restrict__ x,
    const float* __restrict__ w_lin, const float* __restrict__ b_lin,
    const float* __restrict__ g_z, const float* __restrict__ be_z,
    float* __restrict__ out) {
  __shared__ float wtab[16 * CZ];        // wtab[b][c] = w_lin[c,b]+b_lin[c]+be_z[c]; b=15 -> b_lin+be_z
#if USE_ASYNC
  __shared__ float zstage[8 * 2 * CZ];   // per-wave double buffer (512 B slots)
#endif
  const int tid  = threadIdx.x;
  const int lane = tid & 31;
  const int wav  = tid >> 5;
  const int i    = blockIdx.x;

  for (int idx = tid; idx < 16 * CZ; idx += 256) {
    const int b = idx >> 7;
    const int c = idx & (CZ - 1);
    float v = b_lin[c] + be_z[c];
    if (b < NBINS) v += w_lin[c * NBINS + b];
    wtab[idx] = v;
  }

  const float xi0 = x[i * 3 + 0];
  const float xi1 = x[i * 3 + 1];
  const float xi2 = x[i * 3 + 2];
  const float4 gz = reinterpret_cast<const float4*>(g_z)[lane];
  __syncthreads();

  const float* zbase = z   + (size_t)i * N_RES * CZ;
  float*       obase = out + (size_t)i * N_RES * CZ;

  auto process = [&](int j, float4 v) {
    // ---- bin via uniform-grid arithmetic + exact strict-inequality fixup ----
    const float dx = xi0 - x[j * 3 + 0];
    const float dy = xi1 - x[j * 3 + 1];
    const float dz = xi2 - x[j * 3 + 2];
    const float d2 = dx * dx + dy * dy + dz * dz;
    const float d  = sqrtf(d2);
    int p = (int)floorf((d - 3.25f) * 0.8f);
    p = p < 0 ? 0 : (p > 14 ? 14 : p);
    const float e  = 3.25f + 1.25f * (float)p;   // true bin edges (exact f32)
    const float lo = e * e;
    const float e1 = e + 1.25f;
    const float hi = (p == 14) ? INF_F : e1 * e1;
    if (!(d2 > lo))      p -= 1;   // overshoot, or below first edge -> none
    else if (!(d2 < hi)) p += 1;   // undershoot (p<14 here since hi(14)=INF)
    const int bin = (p < 0) ? NBINS : p;

    // ---- wave layernorm over 128 channels (4 per lane) ----
    float s  = v.x + v.y + v.z + v.w;
    float ss = v.x*v.x + v.y*v.y + v.z*v.z + v.w*v.w;
    s = wred(s); ss = wred(ss);
    const float mu  = s * (1.0f / CZ);
    const float var = ss * (1.0f / CZ) - mu * mu;
    const float inv = rsqrtf(var + LN_EPS);
    const float4 wt = reinterpret_cast<const float4*>(&wtab[bin * CZ])[lane];
    const float nmu = -mu;
    float4 ig, o;
    ig.x = inv * gz.x; ig.y = inv * gz.y; ig.z = inv * gz.z; ig.w = inv * gz.w;
    o.x = v.x * ig.x + (nmu * ig.x + wt.x);
    o.y = v.y * ig.y + (nmu * ig.y + wt.y);
    o.z = v.z * ig.z + (nmu * ig.z + wt.z);
    o.w = v.w * ig.w + (nmu * ig.w + wt.w);
    reinterpret_cast<float4*>(obase + (size_t)j * CZ)[lane] = o;
  };

#if USE_ASYNC
  // Double-buffered async global->LDS streaming; constant trip count so the
  // s_wait_asynccnt immediates are compile-time constants.
  constexpr int TRIPS = N_RES / 8;  // 96 rows per wave
  float* slot0 = &zstage[(wav * 2 + 0) * CZ];
  float* slot1 = &zstage[(wav * 2 + 1) * CZ];
  float* slots[2] = { slot0, slot1 };
  async_copy16(zbase + (size_t)wav * CZ + lane * 4, slot0 + lane * 4);
  int s = 0;
#pragma unroll 2
  for (int t = 0; t < TRIPS - 1; ++t) {
    const int j  = wav + t * 8;
    const int jn = j + 8;
    async_copy16(zbase + (size_t)jn * CZ + lane * 4, slots[s ^ 1] + lane * 4);
    __builtin_amdgcn_s_wait_asynccnt(1);   // current row landed in LDS
    const float4 v = reinterpret_cast<const float4*>(slots[s])[lane];
    process(j, v);
    s ^= 1;
  }
  __builtin_amdgcn_s_wait_asynccnt(0);
  {
    const float4 v = reinterpret_cast<const float4*>(slots[s])[lane];
    process(wav + (TRIPS - 1) * 8, v);
  }
#else
  for (int j = wav; j < N_RES; j += 8) {
    if (j + 8 < N_RES)
      __builtin_prefetch(zbase + (size_t)(j + 8) * CZ + lane * 4, 0, 1);
    const float4 v =
        reinterpret_cast<const float4*>(zbase + (size_t)j * CZ)[lane];
    process(j, v);
  }
#endif
}

extern "C" void kernel_launch(void* const* d_in, const int* in_sizes, int n_in,
                              void* d_out, int out_size, void* d_ws, size_t ws_size,
                              hipStream_t stream) {
  (void)in_sizes; (void)n_in; (void)d_ws; (void)ws_size; (void)out_size;
  const float* m    = (const float*)d_in[0];
  const float* z    = (const float*)d_in[1];
  const float* x    = (const float*)d_in[2];
  const float* wlin = (const float*)d_in[3];
  const float* blin = (const float*)d_in[4];
  const float* gm   = (const float*)d_in[5];
  const float* bem  = (const float*)d_in[6];
  const float* gz   = (const float*)d_in[7];
  const float* bez  = (const float*)d_in[8];

  float* out_m = (float*)d_out;                       // [768, 256]
  float* out_z = out_m + (size_t)N_RES * CM;          // [768, 768, 128]

  k_m_layernorm<<<(N_RES * 32 + 255) / 256, 256, 0, stream>>>(m, gm, bem, out_m);
  k_z_update<<<N_RES, 256, 0, stream>>>(z, x, wlin, blin, gz, bez, out_z);
}